// Distance_40140764348439
// MI455X (gfx1250) — compile-verified
//
#include <hip/hip_runtime.h>

typedef __attribute__((ext_vector_type(2))) float v2f;
typedef __attribute__((ext_vector_type(8))) float v8f;

#define EPS_F 1e-6f
#define PDIM 50
#define DDIM 768
#define MTOT (1024 * 197)       // 201728 flattened patch rows
#define ROWS_PER_WG 256         // 16 waves * 16 rows
#define NWG (MTOT / ROWS_PER_WG) // 788, exact
#define KC 192                  // K-chunk staged in LDS
#define NCHUNK (DDIM / KC)      // 4
#define LSTR 196                // padded LDS row stride (dwords); 196 % 64 == 4 -> conflict-free

// ---------------- alpha[p] = ||a_p||^2 + 2*eps*sum(a_p) + D*eps^2 ----------------
__global__ void alpha_kernel(const float* __restrict__ src, float* __restrict__ alpha) {
    int p = threadIdx.x;
    if (p >= 64) return;
    float sq = 0.f, s = 0.f;
    if (p < PDIM) {
        const float* row = src + (long)p * DDIM;
        for (int d = 0; d < DDIM; ++d) {
            float v = row[d];
            sq = __builtin_fmaf(v, v, sq);
            s += v;
        }
    }
    alpha[p] = sq + 2.0f * EPS_F * s + (float)DDIM * EPS_F * EPS_F;
}

// ---------------- main: WMMA GEMM + fused beta + sqrt-reduce ----------------
__global__ __launch_bounds__(512)
void dist_kernel(const float* __restrict__ src, const float* __restrict__ pe,
                 const float* __restrict__ alpha, float* __restrict__ partials) {
    __shared__ float sA[64 * LSTR];   // ~49 KB A chunk, padded rows
    __shared__ float sAlpha[64];
    __shared__ float sRed[16];

    const int tid  = threadIdx.x;
    const int lane = tid & 31;
    const int wave = tid >> 5;
    const int n    = lane & 15;   // N index (patch row within wave tile) / A row index
    const int hi   = lane >> 4;   // K-half selector per WMMA fragment layout

    if (tid < 64) sAlpha[tid] = alpha[tid];

    const int m0 = blockIdx.x * ROWS_PER_WG + wave * 16;
    const float* brow = pe + (long)(m0 + n) * DDIM + 2 * hi;

    v8f acc0 = {}, acc1 = {}, acc2 = {}, acc3 = {};
    float bsq = 0.f, bs = 0.f;

    for (int kc = 0; kc < NCHUNK; ++kc) {
        __syncthreads();
        // Cooperative load of A chunk: 64 rows x KC cols, zero-padded rows >= PDIM.
        // 64*(KC/4)=3072 float4 slots / 512 threads = 6 each, no divergence.
        for (int i = tid; i < 64 * (KC / 4); i += 512) {
            int p   = i / (KC / 4);
            int kk4 = i % (KC / 4);
            float4 v = {0.f, 0.f, 0.f, 0.f};
            if (p < PDIM)
                v = *(const float4*)(src + (long)p * DDIM + kc * KC + kk4 * 4);
            float* d = &sA[p * LSTR + kk4 * 4];
            d[0] = v.x; d[1] = v.y; d[2] = v.z; d[3] = v.w;
        }
        __syncthreads();

        const float* bc = brow + kc * KC;
        const int kbase = 2 * hi;
#pragma unroll 2
        for (int k = 0; k < KC; k += 4) {
            // B fragment (4x16): lane reads pe[m0+n][k0 + 2*hi + {0,1}]
            v2f b = *(const v2f*)(bc + k);
            bsq = __builtin_fmaf(b.x, b.x, bsq);
            bsq = __builtin_fmaf(b.y, b.y, bsq);
            bs += b.x + b.y;

            const int ka = k + kbase;
            // A fragments (16x4): lane reads A[t*16 + n][k0 + 2*hi + {0,1}] from LDS
            v2f a0 = *(const v2f*)&sA[(0 * 16 + n) * LSTR + ka];
            v2f a1 = *(const v2f*)&sA[(1 * 16 + n) * LSTR + ka];
            v2f a2 = *(const v2f*)&sA[(2 * 16 + n) * LSTR + ka];
            v2f a3 = *(const v2f*)&sA[(3 * 16 + n) * LSTR + ka];

            acc0 = __builtin_amdgcn_wmma_f32_16x16x4_f32(false, a0, false, b, (short)0, acc0, false, false);
            acc1 = __builtin_amdgcn_wmma_f32_16x16x4_f32(false, a1, false, b, (short)0, acc1, false, false);
            acc2 = __builtin_amdgcn_wmma_f32_16x16x4_f32(false, a2, false, b, (short)0, acc2, false, false);
            acc3 = __builtin_amdgcn_wmma_f32_16x16x4_f32(false, a3, false, b, (short)0, acc3, false, false);
        }
    }

    // beta_m = sum(b^2) - 2*eps*sum(b); lane n and lane n+16 each hold half the K range
    float betap = __builtin_fmaf(-2.0f * EPS_F, bs, bsq);
    float beta  = betap + __shfl_xor(betap, 16, 32);

    // Epilogue: d(p,m) = sqrt(alpha[p] + beta_m - 2*C[p,m]); C layout: VGPR r -> row r + 8*hi
    float lsum = 0.f;
#define ACC_TILE(ACC, T)                                                          \
    {                                                                             \
        _Pragma("unroll") for (int r = 0; r < 8; ++r) {                           \
            int p = (T) * 16 + r + 8 * hi;                                        \
            if (p < PDIM) {                                                       \
                float d2 = sAlpha[p] + beta - 2.0f * (ACC)[r];                    \
                lsum += sqrtf(fmaxf(d2, 0.f));                                    \
            }                                                                     \
        }                                                                         \
    }
    ACC_TILE(acc0, 0)
    ACC_TILE(acc1, 1)
    ACC_TILE(acc2, 2)
    ACC_TILE(acc3, 3)
#undef ACC_TILE

    // wave32 butterfly reduce
    lsum += __shfl_xor(lsum, 1, 32);
    lsum += __shfl_xor(lsum, 2, 32);
    lsum += __shfl_xor(lsum, 4, 32);
    lsum += __shfl_xor(lsum, 8, 32);
    lsum += __shfl_xor(lsum, 16, 32);
    if (lane == 0) sRed[wave] = lsum;
    __syncthreads();
    if (tid == 0) {
        float t = 0.f;
        for (int i = 0; i < 16; ++i) t += sRed[i];
        partials[blockIdx.x] = t;
    }
}

// ---------------- deterministic final reduction ----------------
__global__ void reduce_kernel(const float* __restrict__ partials, float* __restrict__ out) {
    __shared__ float s[256];
    float t = 0.f;
    for (int i = threadIdx.x; i < NWG; i += 256) t += partials[i];
    s[threadIdx.x] = t;
    __syncthreads();
    for (int off = 128; off > 0; off >>= 1) {
        if (threadIdx.x < off) s[threadIdx.x] += s[threadIdx.x + off];
        __syncthreads();
    }
    if (threadIdx.x == 0)
        out[0] = s[0] * (1.0f / ((float)PDIM * 197.0f * 1024.0f));
}

extern "C" void kernel_launch(void* const* d_in, const int* in_sizes, int n_in,
                              void* d_out, int out_size, void* d_ws, size_t ws_size,
                              hipStream_t stream) {
    const float* src = (const float*)d_in[0];  // [50, 768]
    const float* pe  = (const float*)d_in[1];  // [1024, 197, 768]
    float* out = (float*)d_out;                // scalar
    float* partials = (float*)d_ws;            // [NWG]
    float* alpha    = partials + 1024;         // [64]

    alpha_kernel<<<1, 64, 0, stream>>>(src, alpha);
    dist_kernel<<<NWG, 512, 0, stream>>>(src, pe, alpha, partials);
    reduce_kernel<<<1, 256, 0, stream>>>(partials, out);
}